// OptimizedMoELayer_24257975287910
// MI455X (gfx1250) — compile-verified
//
#include <hip/hip_runtime.h>
#include <hip/hip_bf16.h>
#include <math.h>

#define D_MODEL   1024
#define D_FF      4096
#define N_EXPERTS 8
#define N_TOKENS  8192
#define CAPACITY  1024                 // N_TOKENS / N_EXPERTS * 1.0
#define NK        (N_TOKENS * 2)       // top-2 assignments
#define LB_WEIGHT 0.01f
#define Z_WEIGHT  0.001f

typedef __attribute__((ext_vector_type(2))) float v2f;
typedef __attribute__((ext_vector_type(8))) float v8f;

// ---- workspace layout (float indices) ----
#define WS_PROBSUM 0                                   // float[8]
#define WS_TPE     8                                   // float[8]
#define WS_ZACC    16                                  // float[1]
#define WS_TOPK_E  32                                  // int[NK]
#define WS_TOPK_P  (WS_TOPK_E + NK)                    // float[NK]
#define WS_EXP_TOK (WS_TOPK_P + NK)                    // int[8*CAPACITY]
#define WS_EXP_GATE (WS_EXP_TOK + N_EXPERTS*CAPACITY)  // float[8*CAPACITY]
#define WS_H       50176                               // float[CAPACITY*D_FF]

// ---- CDNA5 async global->LDS DMA helpers ----
__device__ __forceinline__ unsigned lds_off(const void* p) {
    // generic shared pointer: low 32 bits are the LDS byte offset
    return (unsigned)(uintptr_t)p;
}
__device__ __forceinline__ void async_ld128(unsigned lds, const void* sbase, unsigned voff) {
    asm volatile("global_load_async_to_lds_b128 %0, %1, %2"
                 :: "v"(lds), "v"(voff), "s"(sbase) : "memory");
}
__device__ __forceinline__ void wait_async0() {
    asm volatile("s_wait_asynccnt 0x0" ::: "memory");
}

// ---------------- init ----------------
__global__ void moe_init(float* __restrict__ out, float* __restrict__ ws) {
    size_t idx = (size_t)blockIdx.x * blockDim.x + threadIdx.x;
    size_t ytot = (size_t)N_TOKENS * D_MODEL + 2;
    if (idx < ytot) out[idx] = 0.0f;
    if (idx < N_EXPERTS * CAPACITY) ((int*)(ws + WS_EXP_TOK))[idx] = -1;
    if (idx < 17) ws[idx] = 0.0f;   // probsum[8], tpe[8], zacc
}

// ---------------- router: one wave per token ----------------
__global__ __launch_bounds__(256) void moe_router(const float* __restrict__ x,
                                                  const float* __restrict__ gate_w,
                                                  float* __restrict__ ws) {
    __shared__ float gw[N_EXPERTS * D_MODEL];          // 32 KB
    int tid = threadIdx.x;
    for (int i = tid * 4; i < N_EXPERTS * D_MODEL; i += 256 * 4)
        *(float4*)(gw + i) = *(const float4*)(gate_w + i);
    __syncthreads();

    int wave = tid >> 5, lane = tid & 31;
    int t = blockIdx.x * 8 + wave;
    const float* xr = x + (size_t)t * D_MODEL;

    float acc[N_EXPERTS];
#pragma unroll
    for (int e = 0; e < N_EXPERTS; ++e) acc[e] = 0.0f;
    for (int k = lane; k < D_MODEL; k += 32) {
        float xv = xr[k];
#pragma unroll
        for (int e = 0; e < N_EXPERTS; ++e) acc[e] += xv * gw[e * D_MODEL + k];
    }
#pragma unroll
    for (int e = 0; e < N_EXPERTS; ++e) {
#pragma unroll
        for (int off = 16; off > 0; off >>= 1) acc[e] += __shfl_xor(acc[e], off, 32);
    }
    if (lane == 0) {
        float m = acc[0];
#pragma unroll
        for (int e = 1; e < N_EXPERTS; ++e) m = fmaxf(m, acc[e]);
        float se = 0.0f, pr[N_EXPERTS];
#pragma unroll
        for (int e = 0; e < N_EXPERTS; ++e) { pr[e] = __expf(acc[e] - m); se += pr[e]; }
        float inv = 1.0f / se;
#pragma unroll
        for (int e = 0; e < N_EXPERTS; ++e) atomicAdd(&ws[WS_PROBSUM + e], pr[e] * inv);
        float lse = m + __logf(se);
        atomicAdd(&ws[WS_ZACC], lse * lse);

        // top-2 (ties -> lowest expert index, matching lax.top_k)
        int e1 = 0; float l1 = acc[0];
#pragma unroll
        for (int e = 1; e < N_EXPERTS; ++e) if (acc[e] > l1) { l1 = acc[e]; e1 = e; }
        int e2 = -1; float l2 = -3.4e38f;
#pragma unroll
        for (int e = 0; e < N_EXPERTS; ++e)
            if (e != e1 && acc[e] > l2) { l2 = acc[e]; e2 = e; }
        float p1 = 1.0f / (1.0f + __expf(l2 - l1));
        float p2 = 1.0f - p1;
        int* te = (int*)(ws + WS_TOPK_E);
        te[2 * t] = e1; te[2 * t + 1] = e2;
        ws[WS_TOPK_P + 2 * t] = p1; ws[WS_TOPK_P + 2 * t + 1] = p2;
    }
}

// ---------------- capacity: exact lexsort rank per assignment ----------------
#define RCHUNK 2048
__global__ __launch_bounds__(256) void moe_capacity(float* __restrict__ ws) {
    __shared__ int      se_[RCHUNK];
    __shared__ unsigned sp_[RCHUNK];
    const int*   te = (const int*)(ws + WS_TOPK_E);
    const float* tp = ws + WS_TOPK_P;
    int i = blockIdx.x * 256 + threadIdx.x;
    int ei = te[i];
    float pi = tp[i];
    unsigned pbi = __float_as_uint(pi);   // p > 0: float-order == uint-order
    int rank = 0;
    for (int c = 0; c < NK; c += RCHUNK) {
        __syncthreads();
        for (int j = threadIdx.x; j < RCHUNK; j += 256) {
            se_[j] = te[c + j];
            sp_[j] = __float_as_uint(tp[c + j]);
        }
        __syncthreads();
        for (int j = 0; j < RCHUNK; ++j) {
            int jj = c + j;
            if (se_[j] == ei && (sp_[j] > pbi || (sp_[j] == pbi && jj < i))) rank++;
        }
    }
    if (rank < CAPACITY) {
        ((int*)(ws + WS_EXP_TOK))[ei * CAPACITY + rank] = i >> 1;
        ws[WS_EXP_GATE + ei * CAPACITY + rank] = pi;
        if ((i & 1) == 0) atomicAdd(&ws[WS_TPE + ei], 1.0f);   // kept top-1 counts
    }
}

// ---------------- fp32 WMMA GEMM tiles (async double-buffered) ----------------
#define BM  64
#define BN  128
#define BK  16
#define BKP 20    // pad: conflict-free strided fragment reads, 16B-aligned rows
#define NST1 (D_MODEL / BK)   // 64 stages
#define NST2 (D_FF / BK)      // 256 stages

// H = silu(X_e @ W1^T) * (X_e @ W3^T), rows gathered via expert slot table
__global__ __launch_bounds__(256) void moe_ffn1(const float* __restrict__ x,
                                                const float* __restrict__ w1,
                                                const float* __restrict__ w3,
                                                float* __restrict__ ws, int e) {
    __shared__ float As[2][BM * BKP];     // 10 KB
    __shared__ float B1s[2][BN * BKP];    // 20 KB
    __shared__ float B3s[2][BN * BKP];    // 20 KB
    const int* etok = (const int*)(ws + WS_EXP_TOK) + e * CAPACITY;
    float* H = ws + WS_H;
    const float* W1 = w1 + (size_t)e * D_FF * D_MODEL;
    const float* W3 = w3 + (size_t)e * D_FF * D_MODEL;
    int i0 = blockIdx.x * BM;
    int f0 = blockIdx.y * BN;
    int tid = threadIdx.x, wave = tid >> 5, lane = tid & 31;
    int wm = (wave >> 1) * 16;       // wave's M offset within block
    int wn = (wave & 1) * 64;        // wave's N offset within block

    v8f zero = {0.f,0.f,0.f,0.f,0.f,0.f,0.f,0.f};
    v8f acc1[4], acc3[4];
#pragma unroll
    for (int s = 0; s < 4; ++s) { acc1[s] = zero; acc3[s] = zero; }

    // async staging map: A 4 thr/row x 4 floats, B 2 thr/row x 8 floats
    int ar = tid >> 2, acq = (tid & 3) * 4;
    int br = tid >> 1, bcq = (tid & 1) * 8;
    int atok = etok[i0 + ar]; if (atok < 0) atok = 0;   // garbage rows never stored
    unsigned aoff = (unsigned)((atok * D_MODEL + acq) * 4);
    unsigned boff = (unsigned)(((f0 + br) * D_MODEL + bcq) * 4);
    unsigned aL[2]  = { lds_off(&As[0][ar * BKP + acq]),  lds_off(&As[1][ar * BKP + acq]) };
    unsigned b1L[2] = { lds_off(&B1s[0][br * BKP + bcq]), lds_off(&B1s[1][br * BKP + bcq]) };
    unsigned b3L[2] = { lds_off(&B3s[0][br * BKP + bcq]), lds_off(&B3s[1][br * BKP + bcq]) };

    auto issue = [&](int st, int b) {
        unsigned kb = (unsigned)(st * (BK * 4));
        async_ld128(aL[b],       x,  aoff + kb);
        async_ld128(b1L[b],      W1, boff + kb);
        async_ld128(b1L[b] + 16, W1, boff + kb + 16);
        async_ld128(b3L[b],      W3, boff + kb);
        async_ld128(b3L[b] + 16, W3, boff + kb + 16);
    };

    issue(0, 0);
    int kh = (lane >> 4) * 2, mi = lane & 15;
    for (int st = 0; st < NST1; ++st) {
        int b = st & 1;
        wait_async0();
        __syncthreads();
        if (st + 1 < NST1) issue(st + 1, b ^ 1);
        const float* Ab  = As[b];
        const float* B1b = B1s[b];
        const float* B3b = B3s[b];
#pragma unroll
        for (int kk = 0; kk < BK; kk += 4) {
            v2f af = *(const v2f*)(Ab + (wm + mi) * BKP + kk + kh);
#pragma unroll
            for (int s = 0; s < 4; ++s) {
                int bo = (wn + s * 16 + mi) * BKP + kk + kh;
                v2f b1f = *(const v2f*)(B1b + bo);
                v2f b3f = *(const v2f*)(B3b + bo);
                acc1[s] = __builtin_amdgcn_wmma_f32_16x16x4_f32(false, af, false, b1f,
                                                                (short)0, acc1[s], false, false);
                acc3[s] = __builtin_amdgcn_wmma_f32_16x16x4_f32(false, af, false, b3f,
                                                                (short)0, acc3[s], false, false);
            }
        }
    }
    // epilogue: silu(acc1)*acc3 -> H
    int mh = (lane >> 4) * 8, ni = lane & 15;
#pragma unroll
    for (int s = 0; s < 4; ++s) {
#pragma unroll
        for (int v = 0; v < 8; ++v) {
            int m = wm + v + mh;
            int col = f0 + wn + s * 16 + ni;
            float a = acc1[s][v], b = acc3[s][v];
            float h = (a / (1.0f + __expf(-a))) * b;
            H[(size_t)(i0 + m) * D_FF + col] = h;
        }
    }
}

// Y += gate * (H @ W2^T), scattered back to token rows
__global__ __launch_bounds__(256) void moe_ffn2(const float* __restrict__ w2,
                                                float* __restrict__ y,
                                                float* __restrict__ ws, int e) {
    __shared__ float As[2][BM * BKP];
    __shared__ float Bs[2][BN * BKP];
    const int*   etok  = (const int*)(ws + WS_EXP_TOK) + e * CAPACITY;
    const float* egate = ws + WS_EXP_GATE + e * CAPACITY;
    const float* H = ws + WS_H;
    const float* W2 = w2 + (size_t)e * D_MODEL * D_FF;
    int i0 = blockIdx.x * BM;
    int n0 = blockIdx.y * BN;
    int tid = threadIdx.x, wave = tid >> 5, lane = tid & 31;
    int wm = (wave >> 1) * 16, wn = (wave & 1) * 64;

    v8f zero = {0.f,0.f,0.f,0.f,0.f,0.f,0.f,0.f};
    v8f acc[4];
#pragma unroll
    for (int s = 0; s < 4; ++s) acc[s] = zero;

    int ar = tid >> 2, acq = (tid & 3) * 4;
    int br = tid >> 1, bcq = (tid & 1) * 8;
    unsigned aoff = (unsigned)(((i0 + ar) * D_FF + acq) * 4);
    unsigned boff = (unsigned)(((n0 + br) * D_FF + bcq) * 4);
    unsigned aL[2] = { lds_off(&As[0][ar * BKP + acq]), lds_off(&As[1][ar * BKP + acq]) };
    unsigned bL[2] = { lds_off(&Bs[0][br * BKP + bcq]), lds_off(&Bs[1][br * BKP + bcq]) };

    auto issue = [&](int st, int b) {
        unsigned kb = (unsigned)(st * (BK * 4));
        async_ld128(aL[b],      H,  aoff + kb);
        async_ld128(bL[b],      W2, boff + kb);
        async_ld128(bL[b] + 16, W2, boff + kb + 16);
    };

    issue(0, 0);
    int kh = (lane >> 4) * 2, mi = lane & 15;
    for (int st = 0; st < NST2; ++st) {
        int b = st & 1;
        wait_async0();
        __syncthreads();
        if (st + 1 < NST2) issue(st + 1, b ^ 1);
        const float* Ab = As[b];
        const float* Bb = Bs[b];
#pragma unroll
        for (int kk = 0; kk < BK; kk += 4) {
            v2f af = *(const v2f*)(Ab + (wm + mi) * BKP + kk + kh);
#pragma unroll
            for (int s = 0; s < 4; ++s) {
                v2f bf = *(const v2f*)(Bb + (wn + s * 16 + mi) * BKP + kk + kh);
                acc[s] = __builtin_amdgcn_wmma_f32_16x16x4_f32(false, af, false, bf,
                                                               (short)0, acc[s], false, false);
            }
        }
    }
    int mh = (lane >> 4) * 8, ni = lane & 15;
#pragma unroll
    for (int s = 0; s < 4; ++s) {
#pragma unroll
        for (int v = 0; v < 8; ++v) {
            int slot = i0 + wm + v + mh;
            int t = etok[slot];
            if (t >= 0) {
                int col = n0 + wn + s * 16 + ni;
                float g = egate[slot];
                y[(size_t)t * D_MODEL + col] += g * acc[s][v];   // safe: experts serialized
            }
        }
    }
}

// ---------------- losses ----------------
__global__ void moe_losses(float* __restrict__ out, const float* __restrict__ ws) {
    if (blockIdx.x == 0 && threadIdx.x == 0) {
        float lb = 0.0f;
        for (int e = 0; e < N_EXPERTS; ++e) lb += ws[WS_PROBSUM + e] * ws[WS_TPE + e];
        out[(size_t)N_TOKENS * D_MODEL]     = lb * LB_WEIGHT / ((float)N_TOKENS * N_EXPERTS);
        out[(size_t)N_TOKENS * D_MODEL + 1] = ws[WS_ZACC] / (float)N_TOKENS * Z_WEIGHT;
    }
}

extern "C" void kernel_launch(void* const* d_in, const int* in_sizes, int n_in,
                              void* d_out, int out_size, void* d_ws, size_t ws_size,
                              hipStream_t stream) {
    const float* x      = (const float*)d_in[0];
    const float* gate_w = (const float*)d_in[1];
    const float* w1     = (const float*)d_in[2];
    const float* w3     = (const float*)d_in[3];
    const float* w2     = (const float*)d_in[4];
    float* out = (float*)d_out;
    float* ws  = (float*)d_ws;

    size_t ytot = (size_t)N_TOKENS * D_MODEL + 2;
    moe_init<<<(unsigned)((ytot + 255) / 256), 256, 0, stream>>>(out, ws);
    moe_router<<<N_TOKENS / 8, 256, 0, stream>>>(x, gate_w, ws);
    moe_capacity<<<NK / 256, 256, 0, stream>>>(ws);
    for (int e = 0; e < N_EXPERTS; ++e) {
        moe_ffn1<<<dim3(CAPACITY / BM, D_FF / BN), 256, 0, stream>>>(x, w1, w3, ws, e);
        moe_ffn2<<<dim3(CAPACITY / BM, D_MODEL / BN), 256, 0, stream>>>(w2, out, ws, e);
    }
    moe_losses<<<1, 64, 0, stream>>>(out, ws);
}